// SpikeLayer_40759239639843
// MI455X (gfx1250) — compile-verified
//
#include <hip/hip_runtime.h>

// SNN timestep: element-wise over N = 32*128*64*64 floats + global max reduction.
// Memory-bound (~872 MB @ 23.3 TB/s => ~37us floor). Strategy: B128 nontemporal
// vector loads/stores on all 13 streams, global_prefetch_b8 pipelining of the
// next chunk, wave32 shuffle reduction, low-contention encoded-uint atomicMax
// into 64 cache-line-spaced workspace cells.

typedef float v4f __attribute__((ext_vector_type(4)));

#define V_THRESH   1.0f
#define TAU_REFRAC 0.1f
#define DT         1.0f
#define NCELLS      64
#define CELL_STRIDE 16   // 16 uints = 64 bytes -> one cell per cache line

// Monotone (order-preserving) float <-> uint mapping for integer atomicMax.
__device__ __forceinline__ unsigned enc_f32(float f) {
  unsigned u = __float_as_uint(f);
  return (u & 0x80000000u) ? ~u : (u | 0x80000000u);
}
__device__ __forceinline__ float dec_f32(unsigned e) {
  unsigned u = (e & 0x80000000u) ? (e ^ 0x80000000u) : ~e;
  return __uint_as_float(u);
}

__global__ void snn_init_ws(unsigned* __restrict__ ws) {
  if (threadIdx.x < NCELLS) ws[threadIdx.x * CELL_STRIDE] = 0u;  // 0 = minimum key
}

__global__ __launch_bounds__(256)
void snn_step(const float* __restrict__ impulse,
              const float* __restrict__ mem,
              const float* __restrict__ refrac,
              const float* __restrict__ payloads,
              const float* __restrict__ psum,
              const float* __restrict__ scnt,
              const float* __restrict__ timep,
              float* __restrict__ out,
              unsigned* __restrict__ ws,
              int N, int n4)
{
  const float t     = timep[0];
  const float t_ref = t + TAU_REFRAC;

  const v4f* __restrict__ imp4 = (const v4f*)impulse;
  const v4f* __restrict__ mem4 = (const v4f*)mem;
  const v4f* __restrict__ ref4 = (const v4f*)refrac;
  const v4f* __restrict__ pay4 = (const v4f*)payloads;
  const v4f* __restrict__ ps4  = (const v4f*)psum;
  const v4f* __restrict__ sc4  = (const v4f*)scnt;

  // Output streams, reference return order.
  v4f* __restrict__ o_spk = (v4f*)(out + (size_t)0 * N);
  v4f* __restrict__ o_mem = (v4f*)(out + (size_t)1 * N);
  v4f* __restrict__ o_ref = (v4f*)(out + (size_t)2 * N);
  v4f* __restrict__ o_pay = (v4f*)(out + (size_t)3 * N);
  v4f* __restrict__ o_ps  = (v4f*)(out + (size_t)4 * N);
  v4f* __restrict__ o_sc  = (v4f*)(out + (size_t)5 * N);
  v4f* __restrict__ o_st  = (v4f*)(out + (size_t)6 * N);

  float lmax = -__builtin_inff();
  const int stride = (int)(gridDim.x * blockDim.x);

  for (int i = (int)(blockIdx.x * blockDim.x + threadIdx.x); i < n4; i += stride) {
    // Issue all 6 NT loads back-to-back (clause-friendly, single loadcnt ramp).
    v4f vi = __builtin_nontemporal_load(imp4 + i);
    v4f vm = __builtin_nontemporal_load(mem4 + i);
    v4f vr = __builtin_nontemporal_load(ref4 + i);
    v4f vp = __builtin_nontemporal_load(pay4 + i);
    v4f vs = __builtin_nontemporal_load(ps4  + i);
    v4f vc = __builtin_nontemporal_load(sc4  + i);

    // Software pipeline: prefetch next grid-stride chunk into cache
    // (global_prefetch_b8 — no counter pressure, no VGPR return).
    const int ip = i + stride;
    if (ip < n4) {
      __builtin_prefetch((const void*)(imp4 + ip), 0, 0);
      __builtin_prefetch((const void*)(mem4 + ip), 0, 0);
      __builtin_prefetch((const void*)(ref4 + ip), 0, 0);
      __builtin_prefetch((const void*)(pay4 + ip), 0, 0);
      __builtin_prefetch((const void*)(ps4  + ip), 0, 0);
      __builtin_prefetch((const void*)(sc4  + ip), 0, 0);
    }

    v4f r_spk, r_mem, r_ref, r_pay, r_ps, r_sc, r_st;
#pragma unroll
    for (int j = 0; j < 4; ++j) {
      float mi    = (vr[j] > t) ? 0.0f : vi[j];   // refractory gating
      float nm    = vm[j] + mi;                   // new membrane
      bool  sp    = nm >= V_THRESH;               // spike?
      float os    = sp ? V_THRESH : 0.0f;         // output spike
      float resid = nm - V_THRESH;                // residual
      float po    = resid - vs[j];                // payload if spiked

      r_spk[j] = os;
      r_mem[j] = nm - os;                         // reset by subtraction (neg branch never fires)
      r_ref[j] = sp ? t_ref : vr[j];
      r_pay[j] = sp ? po : vp[j];
      r_ps[j]  = sp ? (vs[j] + po) : vs[j];       // faithful rounding: psum + payloads_out
      float sc2 = vc[j] + os;                     // spikecounts + spiked (V_THRESH==1)
      r_sc[j]  = sc2;
      r_st[j]  = sp ? t : 0.0f;                   // spiketrain = t * spiked
      lmax = fmaxf(lmax, sc2);
    }

    __builtin_nontemporal_store(r_spk, o_spk + i);
    __builtin_nontemporal_store(r_mem, o_mem + i);
    __builtin_nontemporal_store(r_ref, o_ref + i);
    __builtin_nontemporal_store(r_pay, o_pay + i);
    __builtin_nontemporal_store(r_ps,  o_ps  + i);
    __builtin_nontemporal_store(r_sc,  o_sc  + i);
    __builtin_nontemporal_store(r_st,  o_st  + i);
  }

  // Scalar tail (N not divisible by 4): handled by block 0 only.
  if (blockIdx.x == 0) {
    float* o = out;
    for (int k = n4 * 4 + (int)threadIdx.x; k < N; k += (int)blockDim.x) {
      float mi    = (refrac[k] > t) ? 0.0f : impulse[k];
      float nm    = mem[k] + mi;
      bool  sp    = nm >= V_THRESH;
      float os    = sp ? V_THRESH : 0.0f;
      float resid = nm - V_THRESH;
      float po    = resid - psum[k];
      o[(size_t)0 * N + k] = os;
      o[(size_t)1 * N + k] = nm - os;
      o[(size_t)2 * N + k] = sp ? t_ref : refrac[k];
      o[(size_t)3 * N + k] = sp ? po : payloads[k];
      o[(size_t)4 * N + k] = sp ? (psum[k] + po) : psum[k];
      float sc2 = scnt[k] + os;
      o[(size_t)5 * N + k] = sc2;
      o[(size_t)6 * N + k] = sp ? t : 0.0f;
      lmax = fmaxf(lmax, sc2);
    }
  }

  // --- block-level max reduction (wave32) ---
  unsigned key = enc_f32(lmax);
#pragma unroll
  for (int off = 16; off > 0; off >>= 1) {
    unsigned o = (unsigned)__shfl_xor((int)key, off, 32);
    if (o > key) key = o;
  }
  __shared__ unsigned smax[8];  // 256 threads = 8 wave32s
  const int lane = (int)(threadIdx.x & 31u);
  const int wid  = (int)(threadIdx.x >> 5u);
  if (lane == 0) smax[wid] = key;
  __syncthreads();
  if (wid == 0) {
    unsigned k2 = (lane < 8) ? smax[lane] : 0u;
#pragma unroll
    for (int off = 4; off > 0; off >>= 1) {
      unsigned o = (unsigned)__shfl_xor((int)k2, off, 32);
      if (o > k2) k2 = o;
    }
    if (lane == 0)
      atomicMax(&ws[(blockIdx.x & (NCELLS - 1)) * CELL_STRIDE], k2);
  }
}

__global__ void snn_finish(const unsigned* __restrict__ ws,
                           const float* __restrict__ timep,
                           float* __restrict__ out, long long off7N)
{
  unsigned key = 0u;
  for (int c = (int)threadIdx.x; c < NCELLS; c += 32) {
    unsigned v = ws[c * CELL_STRIDE];
    if (v > key) key = v;
  }
#pragma unroll
  for (int off = 16; off > 0; off >>= 1) {
    unsigned o = (unsigned)__shfl_xor((int)key, off, 32);
    if (o > key) key = o;
  }
  if (threadIdx.x == 0) {
    float m = dec_f32(key);
    out[off7N] = m * DT / timep[0];
  }
}

extern "C" void kernel_launch(void* const* d_in, const int* in_sizes, int n_in,
                              void* d_out, int out_size, void* d_ws, size_t ws_size,
                              hipStream_t stream) {
  (void)n_in; (void)out_size; (void)ws_size;
  const float* impulse  = (const float*)d_in[0];
  const float* mem      = (const float*)d_in[1];
  const float* refrac   = (const float*)d_in[2];
  const float* payloads = (const float*)d_in[3];
  const float* psum     = (const float*)d_in[4];
  const float* scnt     = (const float*)d_in[5];
  const float* timep    = (const float*)d_in[6];
  float*    out = (float*)d_out;
  unsigned* ws  = (unsigned*)d_ws;

  const int N  = in_sizes[0];
  const int n4 = N / 4;

  snn_init_ws<<<1, 64, 0, stream>>>(ws);

  const int threads = 256;
  int blocks = 4096;                           // cap atomics at 4096 across 64 cells
  int need = (n4 + threads - 1) / threads;
  if (need < 1) need = 1;
  if (blocks > need) blocks = need;

  snn_step<<<blocks, threads, 0, stream>>>(impulse, mem, refrac, payloads, psum,
                                           scnt, timep, out, ws, N, n4);
  snn_finish<<<1, 32, 0, stream>>>(ws, timep, out, (long long)7 * (long long)N);
}